// Model1_77695958385091
// MI455X (gfx1250) — compile-verified
//
#include <hip/hip_runtime.h>
#include <stdint.h>
#include <math.h>

// ---------------------------------------------------------------------------
// Qutrit circuit -> exact bond-9 MPS evaluation.
//   amplitude[h] = sum_{f,d} prod_q U3_q[h_q-h_{q-1}, f_q] *
//                            U2_q[f_q-f_{q-1}, d_q] * U1_q[d_q-d_{q-1}, 0]
// State per prefix: v in C^9 (bond = (f,d)).  Site step = 2x (3x3 contraction)
// + diagonal U3[g,f] scale, g = (h_q - h_{q-1}) mod 3.
// Pipeline: k_setup (unitaries + leaf GEMM panel) -> k_expand (levels 0..6 of
// the prefix tree, single block) -> k_main (chain sites 7..11 per lane, DFS
// sites 12..13 into LDS (node-major), site 14 as wave-wide f32 WMMA GEMMs
// against a 9x9 "hypothesis" panel -> |amp|^2, staged + coalesced NT store).
// ---------------------------------------------------------------------------

#define POW3_12 531441u
#define POW3_15 14348907u

#define TAB_OFF  0      // 15 sites * 42 floats (u1col:6, U2:18, U3:18)
#define A14_OFF  640    // leaf panel: 9x9 complex = 162 floats
#define LVLA_OFF 1024   // prefix-tree level buffer A: 2187*18 floats
#define LVLB_OFF 40448  // prefix-tree level buffer B

#define SUB3(a,b) (((a)-(b)+3)%3)

struct GateDesc { unsigned char g[45]; };

struct cf { float r, i; };
__device__ __forceinline__ cf cmul(cf a, cf b){ cf o; o.r=a.r*b.r-a.i*b.i; o.i=a.r*b.i+a.i*b.r; return o; }
__device__ __forceinline__ cf cmac(cf acc, cf a, cf b){ acc.r+=a.r*b.r-a.i*b.i; acc.i+=a.r*b.i+a.i*b.r; return acc; }

// steps 1+2 of a site: w[f',d] = sum_{d'} v[f',d']*u1[(d-d')%3];
//                      x[f,d]  = sum_{f'} w[f',d]*U2[(f-f')%3,d]
__device__ __forceinline__ void site_uv(const float* __restrict__ tab, const cf* v, cf* x){
  cf u1[3], U2[3][3];
  #pragma unroll
  for (int i=0;i<3;i++){ u1[i].r=tab[i*2]; u1[i].i=tab[i*2+1]; }
  #pragma unroll
  for (int e=0;e<3;e++)
    #pragma unroll
    for (int d=0;d<3;d++){ int o=6+(e*3+d)*2; U2[e][d].r=tab[o]; U2[e][d].i=tab[o+1]; }
  cf w[9];
  #pragma unroll
  for (int fp=0;fp<3;fp++)
    #pragma unroll
    for (int d=0;d<3;d++){
      cf acc; acc.r=0.f; acc.i=0.f;
      #pragma unroll
      for (int dp=0;dp<3;dp++) acc = cmac(acc, v[fp*3+dp], u1[SUB3(d,dp)]);
      w[fp*3+d]=acc;
    }
  #pragma unroll
  for (int f=0;f<3;f++)
    #pragma unroll
    for (int d=0;d<3;d++){
      cf acc; acc.r=0.f; acc.i=0.f;
      #pragma unroll
      for (int fp=0;fp<3;fp++) acc = cmac(acc, w[fp*3+d], U2[SUB3(f,fp)][d]);
      x[f*3+d]=acc;
    }
}

// step 3: v'[f,d] = x[f,d] * U3[g,f]
__device__ __forceinline__ void site_scale(const float* __restrict__ tab, int g, const cf* x, cf* vo){
  cf U3r[3];
  #pragma unroll
  for (int f=0;f<3;f++){ int o=24+(g*3+f)*2; U3r[f].r=tab[o]; U3r[f].i=tab[o+1]; }
  #pragma unroll
  for (int f=0;f<3;f++)
    #pragma unroll
    for (int d=0;d<3;d++) vo[f*3+d]=cmul(x[f*3+d], U3r[f]);
}

// ---------------------------------------------------------------------------
// k_setup: build the 45 3x3 unitaries from angles + gate structure, plus the
// leaf panel A14[l=(f',d')][c=3h+p] = sum_{f,d} U3[(p-h)%3,f]U2[(f-f')%3,d]U1[(d-d')%3,0]
// ---------------------------------------------------------------------------
__global__ void k_setup(const float* __restrict__ angles, GateDesc gd, float* __restrict__ ws){
  int t = threadIdx.x;
  if (t < 45){
    int layer = t/15, q = t%15;
    int code = gd.g[t];
    int mdx = code & 3, j = (code>>2)&3, k = (code>>4)&3;
    float th = angles[t]*0.5f;
    cf U[3][3];
    #pragma unroll
    for (int a=0;a<3;a++)
      #pragma unroll
      for (int b=0;b<3;b++){ U[a][b].r=(a==b)?1.f:0.f; U[a][b].i=0.f; }
    if (mdx==2){
      float coef = sqrtf(2.0f/(float)(j*(j+1)));
      float gv[3] = {0.f,0.f,0.f};
      for (int l=0;l<j;l++) gv[l]=coef;
      gv[j] = -(float)j*coef;
      for (int d=0;d<3;d++){ float xx=th*gv[d]; U[d][d].r=cosf(xx); U[d][d].i=-sinf(xx); }
    } else {
      int a=j-1, b=k-1;
      float c=cosf(th), s=sinf(th);
      U[a][a].r=c; U[b][b].r=c;
      if (mdx==0){ U[a][b].i=-s; U[b][a].i=-s; }
      else       { U[a][b].r=-s; U[b][a].r= s; }
    }
    float* base = ws + TAB_OFF + q*42;
    if (layer==0){
      for (int d=0;d<3;d++){ base[d*2]=U[d][0].r; base[d*2+1]=U[d][0].i; }
    } else if (layer==1){
      for (int e=0;e<3;e++) for (int d=0;d<3;d++){ int o=6+(e*3+d)*2; base[o]=U[e][d].r; base[o+1]=U[e][d].i; }
    } else {
      for (int g=0;g<3;g++) for (int f=0;f<3;f++){ int o=24+(g*3+f)*2; base[o]=U[g][f].r; base[o+1]=U[g][f].i; }
    }
  }
  __syncthreads();
  if (t < 81){
    const float* tab = ws + TAB_OFF + 14*42;
    cf u1[3], U2[3][3], U3[3][3];
    for (int i=0;i<3;i++){ u1[i].r=tab[i*2]; u1[i].i=tab[i*2+1]; }
    for (int e=0;e<3;e++) for (int d=0;d<3;d++){ int o=6+(e*3+d)*2; U2[e][d].r=tab[o]; U2[e][d].i=tab[o+1]; }
    for (int g=0;g<3;g++) for (int f=0;f<3;f++){ int o=24+(g*3+f)*2; U3[g][f].r=tab[o]; U3[g][f].i=tab[o+1]; }
    int l=t/9, c=t%9;
    int fp=l/3, dp=l%3, h=c/3, p=c%3;
    cf acc; acc.r=0.f; acc.i=0.f;
    for (int f=0;f<3;f++)
      for (int d=0;d<3;d++){
        cf term = cmul(U3[SUB3(p,h)][f], U2[SUB3(f,fp)][d]);
        acc = cmac(acc, term, u1[SUB3(d,dp)]);
      }
    ws[A14_OFF+(l*9+c)*2]   = acc.r;
    ws[A14_OFF+(l*9+c)*2+1] = acc.i;
  }
}

// ---------------------------------------------------------------------------
// k_expand: levels 0..6 of the prefix tree (3..2187 states), single block,
// ping-pong through global workspace (syncthreads = block-scope fence).
// ---------------------------------------------------------------------------
__global__ void k_expand(float* __restrict__ ws){
  const int pw[8] = {1,3,9,27,81,243,729,2187};
  for (int L=0; L<=6; L++){
    int N = pw[L+1];
    const float* src = ws + ((L&1)? LVLA_OFF : LVLB_OFF);
    float*       dst = ws + ((L&1)? LVLB_OFF : LVLA_OFF);
    for (int i=threadIdx.x; i<N; i+=blockDim.x){
      cf v[9];
      int p = i%3, par = i/3;
      int hp = (L==0)? 0 : (par%3);
      if (L==0){ for (int z=0;z<9;z++){ v[z].r=0.f; v[z].i=0.f; } v[0].r=1.f; }
      else { const float* sp = src + par*18; for (int z=0;z<9;z++){ v[z].r=sp[z*2]; v[z].i=sp[z*2+1]; } }
      cf x[9];
      site_uv(ws+TAB_OFF+L*42, v, x);
      site_scale(ws+TAB_OFF+L*42, SUB3(p,hp), x, v);
      float* dp2 = dst + i*18;
      for (int z=0;z<9;z++){ dp2[z*2]=v[z].r; dp2[z*2+1]=v[z].i; }
    }
    __syncthreads();
  }
  // level 6 (even) lands in LVLA — read by k_main.
}

// ---------------------------------------------------------------------------
// k_main: 2 waves/block. lane = one 12-digit prefix; chain sites 7..11, DFS
// sites 12..13 -> 9 states/lane staged NODE-MAJOR in LDS (s = n13*32 + lane);
// site 14 = wave-wide WMMA f32 GEMMs; tile t covers states [16t,16t+16) so
// n13 = t>>1 (scalar) and lanePrefix = (t&1)*16 + row — no integer divides
// in the hot loop, and the validity compare is hoisted per tile.
// ---------------------------------------------------------------------------
typedef float v8f __attribute__((ext_vector_type(8)));
typedef float v2f __attribute__((ext_vector_type(2)));

__global__ __launch_bounds__(64) void k_main(const float* __restrict__ ws, float* __restrict__ out){
  __shared__ float s_state[2*288*18];  // 2 waves * 288 level-13 states * C^9
  __shared__ float s_prob[2*864];
  const int lane = threadIdx.x & 31;
  const int wv   = threadIdx.x >> 5;
  const unsigned waveId = blockIdx.x*2u + (unsigned)wv;
  unsigned prefix = waveId*32u + (unsigned)lane;
  unsigned P = (prefix < POW3_12) ? prefix : (POW3_12 - 1u);

  int h[12];
  { unsigned tt=P; for (int q=11;q>=0;q--){ h[q]=(int)(tt%3u); tt/=3u; } }

  const float* lvl6 = ws + LVLA_OFF + (P/243u)*18u;
  __builtin_prefetch(lvl6, 0, 0);          // global_prefetch_b8
  cf v[9];
  #pragma unroll
  for (int i=0;i<9;i++){ v[i].r=lvl6[i*2]; v[i].i=lvl6[i*2+1]; }

  int hprev = h[6];
  #pragma unroll
  for (int q=7;q<=11;q++){
    cf x[9];
    site_uv(ws+TAB_OFF+q*42, v, x);
    site_scale(ws+TAB_OFF+q*42, SUB3(h[q],hprev), x, v);
    hprev = h[q];
  }

  // DFS sites 12..13; store node-major: state s = n13*32 + lane
  float* waveStates = s_state + wv*288*18;
  cf x12[9];
  site_uv(ws+TAB_OFF+12*42, v, x12);
  #pragma unroll
  for (int p12=0;p12<3;p12++){
    cf v12[9];
    site_scale(ws+TAB_OFF+12*42, SUB3(p12,h[11]), x12, v12);
    cf x13[9];
    site_uv(ws+TAB_OFF+13*42, v12, x13);
    #pragma unroll
    for (int p13=0;p13<3;p13++){
      cf v13[9];
      site_scale(ws+TAB_OFF+13*42, SUB3(p13,p12), x13, v13);
      float* st = waveStates + ((p12*3+p13)*32 + lane)*18;
      #pragma unroll
      for (int i=0;i<9;i++){ st[i*2]=v13[i].r; st[i*2+1]=v13[i].i; }
    }
  }
  __syncthreads();

#if defined(__AMDGCN__) && __has_builtin(__builtin_amdgcn_wmma_f32_16x16x4_f32)
  const int half16 = lane >> 4;
  const int col    = lane & 15;
  const int hHyp   = col / 3;            // computed once (col <= 15)
  const int pHyp   = col - hHyp*3;
  const bool colOk = (col < 9);
  // B operands (4x16 chunks of the 9x9 hypothesis panel; cols 9..15 and rows
  // 9..11 zero-padded). Bin = -Bi so Dr = Ar*Br + Ai*(-Bi) without neg bits
  // (NEG[1:0] must be 0 for f32 per ISA 7.12).
  v2f Br[3], Bi[3], Bin[3];
  #pragma unroll
  for (int j=0;j<3;j++){
    int kk = 4*j + half16*2;
    #pragma unroll
    for (int u=0;u<2;u++){
      float re=0.f, im=0.f;
      if (kk+u < 9 && col < 9){
        re = ws[A14_OFF+((kk+u)*9+col)*2];
        im = ws[A14_OFF+((kk+u)*9+col)*2+1];
      }
      Br[j][u]=re; Bi[j][u]=im; Bin[j][u]=-im;
    }
  }
  int n13m3 = 0;                         // n13 % 3, tracked incrementally
  for (int n13=0; n13<9; n13++){
    const bool valid = colOk && (n13m3 == hHyp);
    #pragma unroll
    for (int hb=0; hb<2; hb++){
      const int tile = n13*2 + hb;
      const float* st = waveStates + (tile*16 + col)*18;
      v2f Ar[3], Ai[3];
      #pragma unroll
      for (int j=0;j<3;j++){
        int kk = 4*j + half16*2;
        #pragma unroll
        for (int u=0;u<2;u++){
          float re=0.f, im=0.f;
          if (kk+u < 9){ re = st[(kk+u)*2]; im = st[(kk+u)*2+1]; }
          Ar[j][u]=re; Ai[j][u]=im;
        }
      }
      v8f Dr = {0.f,0.f,0.f,0.f,0.f,0.f,0.f,0.f};
      v8f Di = {0.f,0.f,0.f,0.f,0.f,0.f,0.f,0.f};
      #pragma unroll
      for (int j=0;j<3;j++){
        Dr = __builtin_amdgcn_wmma_f32_16x16x4_f32(false, Ar[j], false, Br[j],  (short)0, Dr, false, false);
        Dr = __builtin_amdgcn_wmma_f32_16x16x4_f32(false, Ai[j], false, Bin[j], (short)0, Dr, false, false);
        Di = __builtin_amdgcn_wmma_f32_16x16x4_f32(false, Ar[j], false, Bi[j],  (short)0, Di, false, false);
        Di = __builtin_amdgcn_wmma_f32_16x16x4_f32(false, Ai[j], false, Br[j],  (short)0, Di, false, false);
      }
      if (valid){
        // lanePrefix = hb*16 + half16*8 + k ; leaf = lanePrefix*27 + n13*3 + pHyp
        const int base2 = wv*864 + (hb*16 + half16*8)*27 + n13*3 + pHyp;
        #pragma unroll
        for (int k=0;k<8;k++){
          float re=Dr[k], im=Di[k];
          s_prob[base2 + k*27] = re*re + im*im;
        }
      }
    }
    n13m3 = (n13m3==2) ? 0 : (n13m3+1);
  }
#else
  // scalar leaf fallback (also taken by the host-side parse pass)
  for (int n=0;n<9;n++){
    const float* st = waveStates + (n*32 + lane)*18;
    int h13 = n%3;
    for (int p=0;p<3;p++){
      cf acc; acc.r=0.f; acc.i=0.f;
      for (int l=0;l<9;l++){
        cf a; a.r=st[l*2]; a.i=st[l*2+1];
        int c = h13*3+p;
        cf b; b.r=ws[A14_OFF+(l*9+c)*2]; b.i=ws[A14_OFF+(l*9+c)*2+1];
        acc = cmac(acc, a, b);
      }
      s_prob[wv*864 + lane*27 + n*3 + p] = acc.r*acc.r + acc.i*acc.i;
    }
  }
#endif
  __syncthreads();
  unsigned base = waveId*864u;
  for (int i=lane; i<864; i+=32){
    unsigned gidx = base + (unsigned)i;
    if (gidx < POW3_15) __builtin_nontemporal_store(s_prob[wv*864+i], &out[gidx]);
  }
}

// ---------------------------------------------------------------------------
// Host: replay numpy RandomState(0) (MT19937, legacy masked-rejection randint
// with 64-bit draws = two 32-bit outputs) to rebuild the circuit structure.
// ---------------------------------------------------------------------------
namespace {
struct MT19937 {
  uint32_t mt[624]; int idx;
  void seed(uint32_t s){
    mt[0]=s;
    for (int i=1;i<624;i++) mt[i]=1812433253u*(mt[i-1]^(mt[i-1]>>30))+(uint32_t)i;
    idx=624;
  }
  uint32_t next(){
    if (idx>=624){
      for (int i=0;i<624;i++){
        uint32_t y=(mt[i]&0x80000000u)|(mt[(i+1)%624]&0x7fffffffu);
        uint32_t v=mt[(i+397)%624]^(y>>1);
        if (y&1u) v^=2567483615u;
        mt[i]=v;
      }
      idx=0;
    }
    uint32_t y=mt[idx++];
    y^=y>>11; y^=(y<<7)&2636928640u; y^=(y<<15)&4022730752u; y^=y>>18;
    return y;
  }
  uint64_t next64(){ uint64_t hi=next(); uint64_t lo=next(); return (hi<<32)|lo; }
  int randint(int n){ // values 0..n-1, masked rejection on 64-bit draws
    uint64_t rng=(uint64_t)(n-1), mask=rng;
    mask|=mask>>1; mask|=mask>>2; mask|=mask>>4; mask|=mask>>8; mask|=mask>>16; mask|=mask>>32;
    for(;;){ uint64_t v=next64()&mask; if (v<=rng) return (int)v; }
  }
};
static void build_gate_desc(GateDesc* gd){
  MT19937 m; m.seed(0u);
  int t=0;
  for (int layer=0; layer<3; layer++){
    for (int q=0; q<15; q++){
      int mdx = m.randint(3);
      int j, k;
      if (mdx==2){ j = m.randint(2)+1; k=0; }
      else { k = m.randint(2)+2; j = m.randint(k-1)+1; }
      gd->g[t++] = (unsigned char)(mdx | (j<<2) | (k<<4));
    }
    // CNOT chain consumes no RNG draws
  }
}
} // namespace

extern "C" void kernel_launch(void* const* d_in, const int* in_sizes, int n_in,
                              void* d_out, int out_size, void* d_ws, size_t ws_size,
                              hipStream_t stream) {
  (void)in_sizes; (void)n_in; (void)out_size; (void)ws_size;
  const float* angles = (const float*)d_in[0];
  float* out = (float*)d_out;
  float* ws  = (float*)d_ws;

  GateDesc gd;
  build_gate_desc(&gd);

  hipLaunchKernelGGL(k_setup,  dim3(1),    dim3(128),  0, stream, angles, gd, ws);
  hipLaunchKernelGGL(k_expand, dim3(1),    dim3(1024), 0, stream, ws);
  // 16608 waves cover 3^12 = 531441 prefixes (32/wave, tail lanes clamped+masked)
  hipLaunchKernelGGL(k_main,   dim3(8304), dim3(64),   0, stream, ws, out);
}